// GATExtractor_89764816486750
// MI455X (gfx1250) — compile-verified
//
#include <hip/hip_runtime.h>
#include <math.h>

// ---------------------------------------------------------------------------
// GAT feature extractor, MI455X (gfx1250, wave32).
//   conv1: h1 = x@W1 (64000x128 -> 64000x41, via v_wmma_f32_16x16x4_f32),
//          edge softmax by dst (float atomics in L2), weighted scatter-sum.
//   conv2: scalar (D=1) version of the same.
// ---------------------------------------------------------------------------

typedef __attribute__((ext_vector_type(2))) float v2f;
typedef __attribute__((ext_vector_type(8))) float v8f;

#define F_IN      128
#define D1        41
#define DPAD      48        // 3 x 16 WMMA column tiles
#define NEG_SLOPE 0.2f

__device__ __forceinline__ float lrelu(float z) { return z > 0.f ? z : NEG_SLOPE * z; }

// float atomic max via monotonic int mapping (maps to global_atomic_{max,min})
__device__ __forceinline__ void atomic_max_f32(float* addr, float v) {
  if (v >= 0.f) atomicMax((int*)addr, __float_as_int(v));
  else          atomicMin((unsigned int*)addr, __float_as_uint(v));
}

// ---------------------------------------------------------------------------
// Kernel 1: h1[N,48] = x[N,128] @ W1[128,41]  (cols 41..47 = 0)
// One wave per 16x16 tile. K loop: 32 x V_WMMA_F32_16X16X4_F32.
// A (16x4 f32): lanes 0-15 hold M=lane K=0,1; lanes 16-31 hold M=lane-16 K=2,3.
// B (4x16 f32): lanes 0-15 hold K=0,1;        lanes 16-31 hold K=2,3 (N=lane&15).
// C (16x16 f32): vgpr r, lane l -> (M = r + 8*(l>>4), N = l&15).
// ---------------------------------------------------------------------------
__global__ void k_gemm_h1(const float* __restrict__ x, const float* __restrict__ W1,
                          float* __restrict__ h1, int N) {
  const int wave  = (int)((blockIdx.x * blockDim.x + threadIdx.x) >> 5);
  const int lane  = (int)(threadIdx.x & 31);
  const int mtile = wave / 3;
  const int ntile = wave % 3;
  if (mtile * 16 >= N) return;                 // uniform per wave: EXEC stays full

  const int l15   = lane & 15;
  const int khalf = (lane >> 4) * 2;           // 0 or 2
  const int col   = ntile * 16 + l15;          // output column (0..47)
  const float bmask = (col < D1) ? 1.f : 0.f;  // zero pad columns, no OOB read
  const int ccol  = col < D1 ? col : (D1 - 1);

  const float* __restrict__ xrow = x + (size_t)(mtile * 16 + l15) * F_IN;

  v8f c = {};
#pragma unroll
  for (int k0 = 0; k0 < F_IN; k0 += 4) {
    __builtin_prefetch(xrow + k0 + 32, 0, 0);  // global_prefetch of the x stream
    v2f a, b;
    a.x = xrow[k0 + khalf];
    a.y = xrow[k0 + khalf + 1];
    b.x = bmask * W1[(k0 + khalf)     * D1 + ccol];
    b.y = bmask * W1[(k0 + khalf + 1) * D1 + ccol];
#if defined(__gfx1250__) && __has_builtin(__builtin_amdgcn_wmma_f32_16x16x4_f32)
    c = __builtin_amdgcn_wmma_f32_16x16x4_f32(false, a, false, b, (short)0, c,
                                              false, false);
#else
    // fallback keeps the TU compiling; histogram will show wmma==0 if taken
    c[0] += a.x * b.x + a.y * b.y;
#endif
  }

#pragma unroll
  for (int r = 0; r < 8; ++r) {
    const int m = mtile * 16 + r + ((lane >> 4) << 3);
    h1[(size_t)m * DPAD + col] = c[r];
  }
}

// ---------------------------------------------------------------------------
// Kernel 2: per-node attention scalars, wave per node.
// ---------------------------------------------------------------------------
__global__ void k_alpha1(const float* __restrict__ h1,
                         const float* __restrict__ a_src, const float* __restrict__ a_dst,
                         float* __restrict__ asrc, float* __restrict__ adst, int N) {
  const int node = (int)((blockIdx.x * blockDim.x + threadIdx.x) >> 5);
  const int lane = (int)(threadIdx.x & 31);
  if (node >= N) return;
  float ps = 0.f, pd = 0.f;
  for (int k = lane; k < D1; k += 32) {
    const float v = h1[(size_t)node * DPAD + k];
    ps = fmaf(v, a_src[k], ps);
    pd = fmaf(v, a_dst[k], pd);
  }
#pragma unroll
  for (int off = 16; off; off >>= 1) {
    ps += __shfl_xor(ps, off, 32);
    pd += __shfl_xor(pd, off, 32);
  }
  if (lane == 0) { asrc[node] = ps; adst[node] = pd; }
}

// ---------------------------------------------------------------------------
// Init: maxima to -inf, sums/numerators to 0.
// ---------------------------------------------------------------------------
__global__ void k_init(float* __restrict__ m1, float* __restrict__ s1,
                       float* __restrict__ num1, float* __restrict__ m2,
                       float* __restrict__ s2, float* __restrict__ num2, int N) {
  const int i = (int)(blockIdx.x * blockDim.x + threadIdx.x);
  if (i < N) {
    m1[i] = -INFINITY; m2[i] = -INFINITY;
    s1[i] = 0.f; s2[i] = 0.f; num2[i] = 0.f;
  }
  if (i < N * DPAD) num1[i] = 0.f;
}

// ---------------------------------------------------------------------------
// Edge pass A: segment max of leaky-relu logits (thread per edge).
// ---------------------------------------------------------------------------
__global__ void k_edge_max(const int* __restrict__ src, const int* __restrict__ dst,
                           const float* __restrict__ asrc, const float* __restrict__ adst,
                           float* __restrict__ m, int E) {
  const int e = (int)(blockIdx.x * blockDim.x + threadIdx.x);
  if (e >= E) return;
  const int s = src[e], d = dst[e];
  atomic_max_f32(&m[d], lrelu(asrc[s] + adst[d]));
}

// ---------------------------------------------------------------------------
// Edge pass B (layer 1): exp-weights + weighted scatter of h1. Wave per edge.
// ---------------------------------------------------------------------------
__global__ void k_edge_agg1(const int* __restrict__ src, const int* __restrict__ dst,
                            const float* __restrict__ asrc, const float* __restrict__ adst,
                            const float* __restrict__ m, const float* __restrict__ h1,
                            float* __restrict__ ssum, float* __restrict__ num, int E) {
  const int e    = (int)((blockIdx.x * blockDim.x + threadIdx.x) >> 5);
  const int lane = (int)(threadIdx.x & 31);
  if (e >= E) return;
  const int s = src[e], d = dst[e];
  const float w = __expf(lrelu(asrc[s] + adst[d]) - m[d]);
  if (lane == 0) atomicAdd(&ssum[d], w);
  if (lane < D1)
    atomicAdd(&num[(size_t)d * DPAD + lane], w * h1[(size_t)s * DPAD + lane]);
  const int k2 = lane + 32;
  if (k2 < D1)
    atomicAdd(&num[(size_t)d * DPAD + k2], w * h1[(size_t)s * DPAD + k2]);
}

// ---------------------------------------------------------------------------
// Node pass: finalize layer 1 (softmax divide, +b1, ReLU) fused with the
// layer-2 projection h2 = relu(out1)@W2 and layer-2 attention scalars.
// Wave per node.
// ---------------------------------------------------------------------------
__global__ void k_node_h2(const float* __restrict__ num, const float* __restrict__ ssum,
                          const float* __restrict__ b1, const float* __restrict__ W2,
                          const float* __restrict__ a2_src, const float* __restrict__ a2_dst,
                          float* __restrict__ h2, float* __restrict__ asrc2,
                          float* __restrict__ adst2, int N) {
  const int node = (int)((blockIdx.x * blockDim.x + threadIdx.x) >> 5);
  const int lane = (int)(threadIdx.x & 31);
  if (node >= N) return;
  const float inv = 1.f / ssum[node];
  float p = 0.f;
  for (int k = lane; k < D1; k += 32) {
    float v = fmaf(num[(size_t)node * DPAD + k], inv, b1[k]);
    v = fmaxf(v, 0.f);                 // ReLU between the two convs
    p = fmaf(v, W2[k], p);
  }
#pragma unroll
  for (int off = 16; off; off >>= 1) p += __shfl_xor(p, off, 32);
  if (lane == 0) {
    h2[node]    = p;
    asrc2[node] = p * a2_src[0];
    adst2[node] = p * a2_dst[0];
  }
}

// ---------------------------------------------------------------------------
// Edge pass B (layer 2): scalar aggregation, thread per edge.
// ---------------------------------------------------------------------------
__global__ void k_edge_agg2(const int* __restrict__ src, const int* __restrict__ dst,
                            const float* __restrict__ asrc, const float* __restrict__ adst,
                            const float* __restrict__ m, const float* __restrict__ h2,
                            float* __restrict__ ssum, float* __restrict__ num, int E) {
  const int e = (int)(blockIdx.x * blockDim.x + threadIdx.x);
  if (e >= E) return;
  const int s = src[e], d = dst[e];
  const float w = __expf(lrelu(asrc[s] + adst[d]) - m[d]);
  atomicAdd(&ssum[d], w);
  atomicAdd(&num[d], w * h2[s]);
}

// ---------------------------------------------------------------------------
// Final: out[n] = relu(num2[n]/s2[n] + b2)
// ---------------------------------------------------------------------------
__global__ void k_final(const float* __restrict__ num2, const float* __restrict__ s2,
                        const float* __restrict__ b2, float* __restrict__ out, int N) {
  const int i = (int)(blockIdx.x * blockDim.x + threadIdx.x);
  if (i >= N) return;
  out[i] = fmaxf(num2[i] / s2[i] + b2[0], 0.f);
}

// ---------------------------------------------------------------------------
extern "C" void kernel_launch(void* const* d_in, const int* in_sizes, int n_in,
                              void* d_out, int out_size, void* d_ws, size_t ws_size,
                              hipStream_t stream) {
  const float* x   = (const float*)d_in[0];
  const int*   ei  = (const int*)d_in[1];
  const float* W1  = (const float*)d_in[2];
  const float* a1s = (const float*)d_in[3];
  const float* a1d = (const float*)d_in[4];
  const float* b1  = (const float*)d_in[5];
  const float* W2  = (const float*)d_in[6];
  const float* a2s = (const float*)d_in[7];
  const float* a2d = (const float*)d_in[8];
  const float* b2  = (const float*)d_in[9];

  const int N = in_sizes[0] / F_IN;
  const int E = in_sizes[1] / 2;
  const int* src = ei;
  const int* dst = ei + E;

  // Workspace carve-up (floats): ~106*N -> ~27 MB for N=64000.
  float* ws    = (float*)d_ws;
  float* h1    = ws; ws += (size_t)N * DPAD;
  float* asrc1 = ws; ws += N;
  float* adst1 = ws; ws += N;
  float* m1    = ws; ws += N;
  float* s1    = ws; ws += N;
  float* num1  = ws; ws += (size_t)N * DPAD;
  float* h2    = ws; ws += N;
  float* asrc2 = ws; ws += N;
  float* adst2 = ws; ws += N;
  float* m2    = ws; ws += N;
  float* s2    = ws; ws += N;
  float* num2  = ws; ws += N;

  float* out = (float*)d_out;

  const int BLK = 256;  // 8 wave32 waves

  // init accumulators
  k_init<<<(N * DPAD + BLK - 1) / BLK, BLK, 0, stream>>>(m1, s1, num1, m2, s2, num2, N);

  // layer 1 GEMM via WMMA: one wave per 16x16 tile, 3 col tiles
  {
    const int waves = ((N + 15) / 16) * 3;
    k_gemm_h1<<<(waves * 32 + BLK - 1) / BLK, BLK, 0, stream>>>(x, W1, h1, N);
  }

  // per-node attention scalars (wave per node)
  k_alpha1<<<(N * 32 + BLK - 1) / BLK, BLK, 0, stream>>>(h1, a1s, a1d, asrc1, adst1, N);

  // layer 1 edge passes
  k_edge_max<<<(E + BLK - 1) / BLK, BLK, 0, stream>>>(src, dst, asrc1, adst1, m1, E);
  k_edge_agg1<<<(E * 32 + BLK - 1) / BLK, BLK, 0, stream>>>(src, dst, asrc1, adst1, m1,
                                                            h1, s1, num1, E);

  // finalize layer 1 + project to layer-2 scalar + layer-2 alphas
  k_node_h2<<<(N * 32 + BLK - 1) / BLK, BLK, 0, stream>>>(num1, s1, b1, W2, a2s, a2d,
                                                          h2, asrc2, adst2, N);

  // layer 2 edge passes (scalar)
  k_edge_max<<<(E + BLK - 1) / BLK, BLK, 0, stream>>>(src, dst, asrc2, adst2, m2, E);
  k_edge_agg2<<<(E + BLK - 1) / BLK, BLK, 0, stream>>>(src, dst, asrc2, adst2, m2,
                                                       h2, s2, num2, E);

  // output
  k_final<<<(N + BLK - 1) / BLK, BLK, 0, stream>>>(num2, s2, b2, out, N);
}